// CustomPointScatter_50783693308343
// MI455X (gfx1250) — compile-verified
//
#include <hip/hip_runtime.h>

// ---------------------------------------------------------------------------
// CustomPointScatter for MI455X (gfx1250, wave32)
//
// pillar_feat[p,c] = mean_k point_features[p,k,c]   (40000 x 32 x 64, f32)
// out[b, c, y, x]  = pillar_feat[p, c]  at coords from voxel_coords
//
// Memory-bound (AI ~= 0.13 FLOP/B; ~600 MB traffic -> ~30-50 us at 23.3 TB/s).
// Reduction done with V_WMMA_F32_16X16X4_F32:
//   A = const 1/32 (16x4), B = 4-point x 16-channel tile -> every row of D is
//   the partial channel mean; 8 K-steps x 4 channel groups per pillar.
// Correct for any K packing (sum is K-permutation invariant) and any shared
// N-to-lane striping of B and D.
// ---------------------------------------------------------------------------

#define NPILLARS 40000
#define NPOINTS  32
#define CFEAT    64
#define BATCH    4
#define HH       512
#define WW       512

typedef __attribute__((ext_vector_type(2))) float v2f;
typedef __attribute__((ext_vector_type(4))) float v4f;
typedef __attribute__((ext_vector_type(8))) float v8f;

// ---------------------------------------------------------------------------
// Kernel 1: zero-fill the 256 MB canvas with non-temporal b128 stores.
// ---------------------------------------------------------------------------
__global__ __launch_bounds__(256) void zero_fill_kernel(v4f* __restrict__ out,
                                                        size_t n4) {
  size_t i = (size_t)blockIdx.x * blockDim.x + threadIdx.x;
  if (i < n4) {
    v4f z = {0.0f, 0.0f, 0.0f, 0.0f};
    __builtin_nontemporal_store(z, out + i);
  }
}

// ---------------------------------------------------------------------------
// Kernel 2: one wave per pillar. WMMA f32 reduction over the 32 points, then
// 2 scattered non-temporal stores per lane (64 channels per pillar).
// ---------------------------------------------------------------------------
__global__ __launch_bounds__(256) void pillar_mean_scatter_kernel(
    const float* __restrict__ pf,   // [40000][32][64]
    const int*   __restrict__ vc,   // [40000][4] = (b, z, y, x)
    float*       __restrict__ out)  // [4][64][512][512]
{
  const int lane   = threadIdx.x & 31;
  const int wave   = threadIdx.x >> 5;
  const int pillar = blockIdx.x * 8 + wave;   // 5000 blocks * 8 waves = 40000
  const int lid    = lane & 15;               // N index within tile
  const int half   = lane >> 4;               // which half-wave

  const float* __restrict__ base = pf + (size_t)pillar * (NPOINTS * CFEAT);

  // A-matrix: every element 1/32 -> D rows are the channel means.
  v2f a;
  a.x = 0.03125f;
  a.y = 0.03125f;

  v8f acc0 = {};
  v8f acc1 = {};
  v8f acc2 = {};
  v8f acc3 = {};

#pragma unroll
  for (int k0 = 0; k0 < NPOINTS; k0 += 4) {
    // Lanes 0-15 carry point rows k0+0 / k0+1, lanes 16-31 rows k0+2 / k0+3.
    const float* __restrict__ r0 = base + (size_t)(k0 + half * 2) * CFEAT + lid;
    const float* __restrict__ r1 = r0 + CFEAT;

    v2f b0, b1, b2, b3;
    b0.x = r0[0];   b0.y = r1[0];    // channels  0..15
    b1.x = r0[16];  b1.y = r1[16];   // channels 16..31
    b2.x = r0[32];  b2.y = r1[32];   // channels 32..47
    b3.x = r0[48];  b3.y = r1[48];   // channels 48..63

    acc0 = __builtin_amdgcn_wmma_f32_16x16x4_f32(false, a, false, b0, (short)0,
                                                 acc0, false, false);
    acc1 = __builtin_amdgcn_wmma_f32_16x16x4_f32(false, a, false, b1, (short)0,
                                                 acc1, false, false);
    acc2 = __builtin_amdgcn_wmma_f32_16x16x4_f32(false, a, false, b2, (short)0,
                                                 acc2, false, false);
    acc3 = __builtin_amdgcn_wmma_f32_16x16x4_f32(false, a, false, b3, (short)0,
                                                 acc3, false, false);
  }

  // Coordinates for this pillar.
  const int b = vc[pillar * 4 + 0];
  const int y = vc[pillar * 4 + 2];
  const int x = vc[pillar * 4 + 3];

  const size_t plane = (size_t)HH * WW;                       // 262144
  const size_t obase = (size_t)b * CFEAT * plane + (size_t)y * WW + (size_t)x;

  // D VGPR0 (element 0 of the v8f): lanes 0-15 hold row M=0, lanes 16-31 row
  // M=8 — all rows identical, so every lane holds the mean of the channel it
  // loaded: cg*16 + (lane & 15).
  float m0, m1;
  int   c0, c1;
  if (half == 0) {
    m0 = acc0[0]; c0 = lid;        // channels  0..15
    m1 = acc1[0]; c1 = 16 + lid;   // channels 16..31
  } else {
    m0 = acc2[0]; c0 = 32 + lid;   // channels 32..47
    m1 = acc3[0]; c1 = 48 + lid;   // channels 48..63
  }
  __builtin_nontemporal_store(m0, out + obase + (size_t)c0 * plane);
  __builtin_nontemporal_store(m1, out + obase + (size_t)c1 * plane);
}

// ---------------------------------------------------------------------------
// Launch
// ---------------------------------------------------------------------------
extern "C" void kernel_launch(void* const* d_in, const int* in_sizes, int n_in,
                              void* d_out, int out_size, void* d_ws, size_t ws_size,
                              hipStream_t stream) {
  const float* pf = (const float*)d_in[0];  // point_features (40000*32*64 f32)
  const int*   vc = (const int*)d_in[1];    // voxel_coords   (40000*4 i32)
  float*       out = (float*)d_out;         // 4*64*512*512 f32

  // 1) zero the canvas (67,108,864 floats = 16,777,216 x b128)
  const size_t n4 = (size_t)out_size / 4;
  const int    zb = 256;
  const int    zg = (int)((n4 + zb - 1) / zb);  // 65536 blocks
  zero_fill_kernel<<<zg, zb, 0, stream>>>((v4f*)out, n4);

  // 2) mean-reduce + scatter: 8 pillars (waves) per 256-thread block
  pillar_mean_scatter_kernel<<<NPILLARS / 8, 256, 0, stream>>>(pf, vc, out);
}